// Loss_343597383760
// MI455X (gfx1250) — compile-verified
//
#include <hip/hip_runtime.h>
#include <math.h>

// Loss = -sum_i [ l_i*sig(x_i) + (1-l_i)*(1-sig(x_i)) ] / ((1 + #neg) * N)
// x_i gathered from pv[rel, e1, e2]. Gather/HBM-bound (~36-40 MB traffic).

typedef __attribute__((ext_vector_type(2))) float v2f;
typedef __attribute__((ext_vector_type(8))) float v8f;

#define ENT 4096
#define BLOCKS 512
#define THREADS 256

__global__ __launch_bounds__(THREADS) void gather_loss_partial(
    const float* __restrict__ pv,
    const int*   __restrict__ rel,
    const int*   __restrict__ e1,
    const int*   __restrict__ e2,
    const int*   __restrict__ lab,
    float*       __restrict__ partials,
    int N)
{
    const int tid    = blockIdx.x * blockDim.x + threadIdx.x;
    const int stride = gridDim.x * blockDim.x;

    float tsum = 0.0f;   // label-weighted sigmoid terms
    float negc = 0.0f;   // count of label==0 (exact integers in fp32)

    for (int i = tid; i < N; i += stride) {
        const int r = rel[i];
        const int a = e1[i];
        const int b = e2[i];
        const int l = lab[i];
        const size_t off = ((size_t)r * ENT + (size_t)a) * (size_t)ENT + (size_t)b;
        // Random single-touch gather: non-temporal to avoid cache pollution.
        const float x = __builtin_nontemporal_load(pv + off);
        // l==1: sigmoid(x) = 1/(1+e^{-x}); l==0: 1-sigmoid(x) = 1/(1+e^{x})
        const float z    = l ? -x : x;
        const float term = 1.0f / (1.0f + expf(z));
        tsum += term;
        negc += (l == 0) ? 1.0f : 0.0f;
    }

    // wave32 tree reduction (warpSize == 32 on gfx1250)
    #pragma unroll
    for (int off = 16; off > 0; off >>= 1) {
        tsum += __shfl_xor(tsum, off, 32);
        negc += __shfl_xor(negc, off, 32);
    }

    __shared__ float s_t[THREADS / 32];
    __shared__ float s_n[THREADS / 32];
    const int lane = threadIdx.x & 31;
    const int wid  = threadIdx.x >> 5;
    if (lane == 0) { s_t[wid] = tsum; s_n[wid] = negc; }
    __syncthreads();

    if (wid == 0) {
        float t = (lane < (THREADS / 32)) ? s_t[lane] : 0.0f;
        float n = (lane < (THREADS / 32)) ? s_n[lane] : 0.0f;
        #pragma unroll
        for (int off = (THREADS / 64); off > 0; off >>= 1) {
            t += __shfl_xor(t, off, 32);
            n += __shfl_xor(n, off, 32);
        }
        if (lane == 0) {
            partials[2 * blockIdx.x]     = t;
            partials[2 * blockIdx.x + 1] = n;
        }
    }
}

// One wave. Cross-lane sum via V_WMMA_F32_16X16X4_F32 with an all-ones B:
// D[m][n] = sum_k A[m][k], so with A holding one lane value each (a={v,0}),
// each D column equals the pairwise sums x[m]+x[m+16]; summing a lane's 8
// accumulator VGPRs (rows M=0..7 / M=8..15 per the C/D
// layout) + one shfl_xor(16) yields the exact 32-lane total.
__global__ __launch_bounds__(32) void finalize_loss(
    const float* __restrict__ partials,
    int nblocks,
    float* __restrict__ out,
    int N)
{
    const int lane = threadIdx.x;
    float t = 0.0f, n = 0.0f;
    for (int i = lane; i < nblocks; i += 32) {
        t += partials[2 * i];
        n += partials[2 * i + 1];
    }

    v2f a_t;  a_t.x = t;     a_t.y = 0.0f;
    v2f a_n;  a_n.x = n;     a_n.y = 0.0f;
    v2f ones; ones.x = 1.0f; ones.y = 1.0f;
    v8f c0 = {};

    v8f dt = __builtin_amdgcn_wmma_f32_16x16x4_f32(
        false, a_t, false, ones, (short)0, c0, false, false);
    v8f dn = __builtin_amdgcn_wmma_f32_16x16x4_f32(
        false, a_n, false, ones, (short)0, c0, false, false);

    float st = dt[0] + dt[1] + dt[2] + dt[3] + dt[4] + dt[5] + dt[6] + dt[7];
    float sn = dn[0] + dn[1] + dn[2] + dn[3] + dn[4] + dn[5] + dn[6] + dn[7];
    st += __shfl_xor(st, 16, 32);
    sn += __shfl_xor(sn, 16, 32);

    if (lane == 0) {
        out[0] = -st / ((1.0f + sn) * (float)N);
    }
}

extern "C" void kernel_launch(void* const* d_in, const int* in_sizes, int n_in,
                              void* d_out, int out_size, void* d_ws, size_t ws_size,
                              hipStream_t stream) {
    const float* pv  = (const float*)d_in[0];
    const int*   rel = (const int*)d_in[1];
    const int*   e1  = (const int*)d_in[2];
    const int*   e2  = (const int*)d_in[3];
    const int*   lab = (const int*)d_in[4];
    const int N = in_sizes[1];

    float* partials = (float*)d_ws;  // 2 * BLOCKS floats (4 KB)

    gather_loss_partial<<<BLOCKS, THREADS, 0, stream>>>(
        pv, rel, e1, e2, lab, partials, N);
    finalize_loss<<<1, 32, 0, stream>>>(
        partials, BLOCKS, (float*)d_out, N);
}